// GraphAttentionLayer_352187318397
// MI455X (gfx1250) — compile-verified
//
#include <hip/hip_runtime.h>
#include <math.h>

// ---------------------------------------------------------------------------
// GAT layer, fused flash-attention style, f32 WMMA (V_WMMA_F32_16X16X4_F32).
//   B=16, T=12 -> BT=192 slices; N=512 nodes; Fin=Fout=64.
// Roofline: ~7.3 GFLOP total, but reference materializes 192x512x512 att
// (201MB). We fuse: ~100MB HBM traffic ~= 4us at 23.3 TB/s. adj (1MB) and
// per-slice Wh tiles (128KB) live in the 192MB L2.
// ---------------------------------------------------------------------------

typedef __attribute__((ext_vector_type(2))) float v2f;
typedef __attribute__((ext_vector_type(8))) float v8f;

#define BT   192
#define NN   512
#define FF   64
#define GAT_ALPHA 0.2f
#define GAT_NEG_INF -9.0e15f

// ---------------------------------------------------------------------------
// Kernel 1: WhT[bt][col][row] = (h[bt] @ W)[row][col]   (store transposed)
// One wave per (bt, 16-row tile): 4 accumulators (16x16), K=64 in steps of 4.
// ---------------------------------------------------------------------------
__global__ __launch_bounds__(32) void gat_wh_kernel(const float* __restrict__ h,
                                                    const float* __restrict__ W,
                                                    float* __restrict__ WhT) {
    const int blk  = blockIdx.x;        // bt*32 + rowtile
    const int bt   = blk >> 5;
    const int n0   = (blk & 31) << 4;
    const int lane = threadIdx.x;
    const int half = lane >> 4;         // 0: K=0,1   1: K=2,3   (A & B frags)
    const int l16  = lane & 15;
    const int koff = half * 2;

    const float* hrow = h + ((size_t)bt * NN + (n0 + l16)) * FF;

    v8f acc0 = {}, acc1 = {}, acc2 = {}, acc3 = {};
    for (int k = 0; k < FF; k += 4) {
        // A fragment: row = n0+l16, K = k+koff, k+koff+1 (contiguous in h)
        v2f a;
        a.x = hrow[k + koff];
        a.y = hrow[k + koff + 1];
        // B fragments: col = c*16 + l16, K rows k+koff, k+koff+1 (stride FF)
        const float* wp = W + (size_t)(k + koff) * FF + l16;
        v2f b0; b0.x = wp[0];  b0.y = wp[FF];
        v2f b1; b1.x = wp[16]; b1.y = wp[16 + FF];
        v2f b2; b2.x = wp[32]; b2.y = wp[32 + FF];
        v2f b3; b3.x = wp[48]; b3.y = wp[48 + FF];
        acc0 = __builtin_amdgcn_wmma_f32_16x16x4_f32(false, a, false, b0, (short)0, acc0, false, false);
        acc1 = __builtin_amdgcn_wmma_f32_16x16x4_f32(false, a, false, b1, (short)0, acc1, false, false);
        acc2 = __builtin_amdgcn_wmma_f32_16x16x4_f32(false, a, false, b2, (short)0, acc2, false, false);
        acc3 = __builtin_amdgcn_wmma_f32_16x16x4_f32(false, a, false, b3, (short)0, acc3, false, false);
    }
    // C/D layout: VGPR r, lane l -> row r + 8*(l>=16), col = l%16 (+c*16).
    // Transposed store: WhT[bt][col][row]; rows r contiguous -> float4 stores.
    v8f accs[4] = {acc0, acc1, acc2, acc3};
    #pragma unroll
    for (int c = 0; c < 4; ++c) {
        float* op = WhT + ((size_t)bt * FF + c * 16 + l16) * NN + n0 + half * 8;
        float4 lo = make_float4(accs[c][0], accs[c][1], accs[c][2], accs[c][3]);
        float4 hi = make_float4(accs[c][4], accs[c][5], accs[c][6], accs[c][7]);
        *(float4*)(op)     = lo;
        *(float4*)(op + 4) = hi;
    }
}

// ---------------------------------------------------------------------------
// Kernel 2: Wh1/Wh2 = Wh @ a1 / a2. One thread per (bt, row); reads WhT with
// lane-contiguous (coalesced) accesses.
// ---------------------------------------------------------------------------
__global__ void gat_proj_kernel(const float* __restrict__ WhT,
                                const float* __restrict__ a,
                                float* __restrict__ Wh1,
                                float* __restrict__ Wh2) {
    const int idx = blockIdx.x * blockDim.x + threadIdx.x; // bt*NN + n
    if (idx >= BT * NN) return;
    const int bt = idx >> 9;
    const int n  = idx & (NN - 1);
    const float* base = WhT + (size_t)bt * FF * NN + n;
    float s1 = 0.f, s2 = 0.f;
    #pragma unroll 8
    for (int k = 0; k < FF; ++k) {
        float v = base[(size_t)k * NN];
        s1 += v * a[k];
        s2 += v * a[FF + k];
    }
    Wh1[idx] = s1;
    Wh2[idx] = s2;
}

// ---------------------------------------------------------------------------
// Kernel 3: fused masked-softmax attention + (att @ Wh) + ELU.
// One wave per (bt, 16-row tile). Pass 1: online softmax stats (2 lanes per
// row). Pass 2: probabilities generated directly in A-fragment layout, 4 f32
// WMMAs per K=4 step (512 WMMAs/block). att is never materialized.
// ---------------------------------------------------------------------------
__global__ __launch_bounds__(32) void gat_attn_kernel(const float* __restrict__ WhT,
                                                      const float* __restrict__ Wh1,
                                                      const float* __restrict__ Wh2,
                                                      const float* __restrict__ adj,
                                                      float* __restrict__ out) {
    const int blk  = blockIdx.x;
    const int bt   = blk >> 5;
    const int n0   = (blk & 31) << 4;
    const int lane = threadIdx.x;
    const int half = lane >> 4;
    const int l16  = lane & 15;
    const int row  = n0 + l16;          // A-fragment row for this lane

    const float  wh1_i = Wh1[bt * NN + row];
    const float* wh2p  = Wh2 + bt * NN;
    const float* adjp  = adj + (size_t)row * NN;

    // ---- Pass 1: row max m and sum s (lane pair (l16, l16+16) splits cols)
    float m = -INFINITY, s = 0.f;
    for (int t = 0; t < NN / 2; ++t) {
        const int j = 2 * t + half;
        float e = wh1_i + wh2p[j];
        e = fmaxf(e, GAT_ALPHA * e);                 // LeakyReLU (alpha<1)
        e = (adjp[j] > 0.f) ? e : GAT_NEG_INF;       // adjacency mask
        const float mn = fmaxf(m, e);
        s = s * __expf(m - mn) + __expf(e - mn);
        m = mn;
    }
    {   // combine the two half-lanes of each row
        const float mo = __shfl_xor(m, 16, 32);
        const float so = __shfl_xor(s, 16, 32);
        const float mn = fmaxf(m, mo);
        s = s * __expf(m - mn) + so * __expf(mo - mn);
        m = mn;
    }
    const float inv_s = 1.0f / s;

    // ---- Pass 2: P(i,j) in A-frag layout, accumulate att @ Wh via WMMA
    v8f acc0 = {}, acc1 = {}, acc2 = {}, acc3 = {};
    const float* whbase = WhT + (size_t)bt * FF * NN;
    const int koff = half * 2;
    for (int j0 = 0; j0 < NN; j0 += 4) {
        const int j = j0 + koff;
        v2f pa;
        #pragma unroll
        for (int u = 0; u < 2; ++u) {
            float e = wh1_i + wh2p[j + u];
            e = fmaxf(e, GAT_ALPHA * e);
            e = (adjp[j + u] > 0.f) ? e : GAT_NEG_INF;
            pa[u] = __expf(e - m) * inv_s;
        }
        // B fragments: Wh rows j, j+1 at col c*16+l16 -> contiguous in WhT
        const float* wp = whbase + (size_t)l16 * NN + j;
        v2f b0 = *(const v2f*)(wp);
        v2f b1 = *(const v2f*)(wp + 16 * NN);
        v2f b2 = *(const v2f*)(wp + 32 * NN);
        v2f b3 = *(const v2f*)(wp + 48 * NN);
        acc0 = __builtin_amdgcn_wmma_f32_16x16x4_f32(false, pa, false, b0, (short)0, acc0, false, false);
        acc1 = __builtin_amdgcn_wmma_f32_16x16x4_f32(false, pa, false, b1, (short)0, acc1, false, false);
        acc2 = __builtin_amdgcn_wmma_f32_16x16x4_f32(false, pa, false, b2, (short)0, acc2, false, false);
        acc3 = __builtin_amdgcn_wmma_f32_16x16x4_f32(false, pa, false, b3, (short)0, acc3, false, false);
    }

    // ---- ELU + store (row-major output, matches reference layout)
    v8f accs[4] = {acc0, acc1, acc2, acc3};
    float* op = out + ((size_t)bt * NN + n0 + half * 8) * FF;
    #pragma unroll
    for (int c = 0; c < 4; ++c) {
        #pragma unroll
        for (int r = 0; r < 8; ++r) {
            float x = accs[c][r];
            x = (x > 0.f) ? x : (__expf(x) - 1.0f);  // ELU (alpha=1)
            op[(size_t)r * FF + c * 16 + l16] = x;
        }
    }
}

// ---------------------------------------------------------------------------
extern "C" void kernel_launch(void* const* d_in, const int* in_sizes, int n_in,
                              void* d_out, int out_size, void* d_ws, size_t ws_size,
                              hipStream_t stream) {
    const float* h   = (const float*)d_in[0];   // (16,12,512,64)
    const float* adj = (const float*)d_in[1];   // (512,512)
    const float* W   = (const float*)d_in[2];   // (64,64)
    const float* a   = (const float*)d_in[3];   // (128,1)
    float* out = (float*)d_out;

    float* WhT = (float*)d_ws;                          // BT*FF*NN floats (~25MB)
    float* Wh1 = WhT + (size_t)BT * FF * NN;            // BT*NN floats
    float* Wh2 = Wh1 + (size_t)BT * NN;                 // BT*NN floats

    const int tiles = BT * (NN / 16);                   // 6144 one-wave blocks
    gat_wh_kernel<<<tiles, 32, 0, stream>>>(h, W, WhT);
    gat_proj_kernel<<<(BT * NN + 255) / 256, 256, 0, stream>>>(WhT, a, Wh1, Wh2);
    gat_attn_kernel<<<tiles, 32, 0, stream>>>(WhT, Wh1, Wh2, adj, out);
}